// GraphAttentionLayer_EFA_37400575214186
// MI455X (gfx1250) — compile-verified
//
#include <hip/hip_runtime.h>
#include <math.h>

#define N 2048
#define D_IN 128
#define D_OUT 32
#define D_EDGE 8
#define ALPHA 0.2f
#define NEG_BIG -9e15f
#define ROWSTRIDE 2050   // 16-row score strip, padded to avoid LDS bank conflicts

typedef __attribute__((ext_vector_type(2))) float v2f;
typedef __attribute__((ext_vector_type(8))) float v8f;

__device__ __forceinline__ v8f wmma_f32_k4(v2f a, v2f b, v8f c) {
  // D = A(16x4, f32) * B(4x16, f32) + C(16x16, f32)
  return __builtin_amdgcn_wmma_f32_16x16x4_f32(false, a, false, b, (short)0, c,
                                               false, false);
}

__device__ __forceinline__ float lrelu(float x) { return x >= 0.f ? x : ALPHA * x; }

// ---------------------------------------------------------------------------
// v1 = Wh1 @ ah, v2 = Wh2 @ ah   (128-vectors; lets s1/s2 be plain matvecs)
// ---------------------------------------------------------------------------
__global__ void prep_vec_kernel(const float* __restrict__ Wh1,
                                const float* __restrict__ Wh2,
                                const float* __restrict__ ah,
                                float* __restrict__ v1, float* __restrict__ v2) {
  const int t = threadIdx.x;  // 0..127
  float a1 = 0.f, a2 = 0.f;
  for (int f = 0; f < D_OUT; ++f) {
    const float w = ah[f];
    a1 += Wh1[t * D_OUT + f] * w;
    a2 += Wh2[t * D_OUT + f] * w;
  }
  v1[t] = a1;
  v2[t] = a2;
}

// s1[i] = input[i,:] . v1 ; s2[i] = input[i,:] . v2
__global__ void prep_s_kernel(const float* __restrict__ input,
                              const float* __restrict__ v1,
                              const float* __restrict__ v2,
                              float* __restrict__ s1, float* __restrict__ s2) {
  const int i = blockIdx.x * blockDim.x + threadIdx.x;
  const float* row = input + (size_t)i * D_IN;
  float a = 0.f, b = 0.f;
  for (int k = 0; k < D_IN; ++k) {
    const float x = row[k];
    a += x * v1[k];
    b += x * v2[k];
  }
  s1[i] = a;
  s2[i] = b;
}

// ---------------------------------------------------------------------------
// h = input (2048x128) @ Wh (128x32) via v_wmma_f32_16x16x4_f32
// One wave per 16-row tile; K=128 in 32 steps of 4; two N-tiles (0..15,16..31)
// ---------------------------------------------------------------------------
__global__ void h_gemm_kernel(const float* __restrict__ input,
                              const float* __restrict__ Wh,
                              float* __restrict__ h) {
  const int lane = threadIdx.x & 31;
  const int i0 = blockIdx.x * 16;
  const int m = lane & 15;              // A-matrix row (M)
  const int kd = (lane >> 4) << 1;      // lanes 16-31 hold K+2,K+3
  const int n = lane & 15;              // B/C column within N-tile
  v8f c0 = {};
  v8f c1 = {};
  for (int k0 = 0; k0 < D_IN; k0 += 4) {
    v2f a = *(const v2f*)(input + (size_t)(i0 + m) * D_IN + k0 + kd);
    v2f b0, b1;
    b0.x = Wh[(k0 + kd) * D_OUT + n];
    b0.y = Wh[(k0 + kd + 1) * D_OUT + n];
    b1.x = Wh[(k0 + kd) * D_OUT + 16 + n];
    b1.y = Wh[(k0 + kd + 1) * D_OUT + 16 + n];
    c0 = wmma_f32_k4(a, b0, c0);
    c1 = wmma_f32_k4(a, b1, c1);
  }
  const int row = (lane >> 4) << 3;  // C layout: VGPR v -> M = v + 8*(lane>=16)
  for (int v = 0; v < 8; ++v) {
    h[(size_t)(i0 + row + v) * D_OUT + n] = c0[v];
    h[(size_t)(i0 + row + v) * D_OUT + 16 + n] = c1[v];
  }
}

// ---------------------------------------------------------------------------
// Fused: scores(edge WMMA) -> mask -> softmax -> h_prime = P @ h (WMMA)
// Block = 16 rows of the attention matrix; 256 threads = 8 waves.
// Dynamic LDS: 16 x ROWSTRIDE score strip + 8 x 512 cross-wave partials.
// ---------------------------------------------------------------------------
__global__ void fused_attention_kernel(const float* __restrict__ edge_feat,
                                       const int* __restrict__ adj,
                                       const float* __restrict__ Wf,
                                       const float* __restrict__ af,
                                       const float* __restrict__ bf,
                                       const float* __restrict__ s1,
                                       const float* __restrict__ s2,
                                       const float* __restrict__ h,
                                       float* __restrict__ h_prime) {
  extern __shared__ float smem[];
  float* sc = smem;                       // 16 * ROWSTRIDE floats
  float* part = smem + 16 * ROWSTRIDE;    // 8 * 512 floats
  __shared__ float s_inv[16];

  const int tid = threadIdx.x;
  const int lane = tid & 31;
  const int wave = tid >> 5;
  const int i0 = blockIdx.x * 16;

  const int nn = lane & 15;
  const int kd = (lane >> 4) << 1;  // 0 or 2

  // B operands from Wf (8x32), resident in registers for the whole stage.
  v2f bw00, bw01, bw10, bw11;
  bw00.x = Wf[(0 + kd) * D_OUT + nn];
  bw00.y = Wf[(1 + kd) * D_OUT + nn];
  bw01.x = Wf[(0 + kd) * D_OUT + 16 + nn];
  bw01.y = Wf[(1 + kd) * D_OUT + 16 + nn];
  bw10.x = Wf[(4 + kd) * D_OUT + nn];
  bw10.y = Wf[(5 + kd) * D_OUT + nn];
  bw11.x = Wf[(4 + kd) * D_OUT + 16 + nn];
  bw11.y = Wf[(5 + kd) * D_OUT + 16 + nn];
  const float af_lo = af[nn], af_hi = af[16 + nn];
  const float bf_lo = bf[nn], bf_hi = bf[16 + nn];
  v8f cini0, cini1;
  for (int v = 0; v < 8; ++v) {
    cini0[v] = bf_lo;   // C seeded with bias bf => f = e@Wf + bf from one WMMA chain
    cini1[v] = bf_hi;
  }

  // ---- Stage A: build masked score strip sc[16][2048] --------------------
  // Task t: row i = i0 + t/128, 16-edge group j0 = (t%128)*16. A-matrix M = j.
  for (int t = wave; t < 16 * 128; t += 8) {
    const int il = t >> 7;
    const int i = i0 + il;
    const int j0 = (t & 127) << 4;
    const float* ep = edge_feat + ((size_t)i * N + (j0 + nn)) * D_EDGE + kd;
    __builtin_prefetch(ep + 8 * 16 * D_EDGE, 0, 1);  // next j-group of this row
    v2f a0 = *(const v2f*)(ep);       // K = kd, kd+1
    v2f a1 = *(const v2f*)(ep + 4);   // K = 4+kd, 5+kd
    v8f f0 = wmma_f32_k4(a0, bw00, cini0);
    f0 = wmma_f32_k4(a1, bw10, f0);
    v8f f1 = wmma_f32_k4(a0, bw01, cini1);
    f1 = wmma_f32_k4(a1, bw11, f1);
    // cp[j] = lrelu( sum_n lrelu(f[j,n]) * af[n] ): per-lane partial over its n
    float p[8];
    for (int v = 0; v < 8; ++v)
      p[v] = lrelu(f0[v]) * af_lo + lrelu(f1[v]) * af_hi;
    // reduce across the 16 lanes of each half (n = 0..15 and 16..31 combined)
    for (int mask = 1; mask <= 8; mask <<= 1)
      for (int v = 0; v < 8; ++v) p[v] += __shfl_xor(p[v], mask, 32);
    if ((lane & 8) == 0) {  // lanes 0-7 -> rows 0-7, lanes 16-23 -> rows 8-15
      const int md = lane & 7;
      float t0 = (md & 1) ? p[1] : p[0];
      float t1 = (md & 1) ? p[3] : p[2];
      float t2 = (md & 1) ? p[5] : p[4];
      float t3 = (md & 1) ? p[7] : p[6];
      float u0 = (md & 2) ? t1 : t0;
      float u1 = (md & 2) ? t3 : t2;
      float cp = (md & 4) ? u1 : u0;
      const int j = j0 + ((lane >> 4) << 3) + md;
      float v = lrelu(s1[i] + s2[j]) + lrelu(cp);
      v = (adj[(size_t)i * N + j] > 0) ? v : NEG_BIG;
      sc[il * ROWSTRIDE + j] = v;
    }
  }
  __syncthreads();

  // ---- Stage B: row softmax; store unnormalized exp, keep 1/sum ----------
  for (int r = wave; r < 16; r += 8) {
    float* row = sc + r * ROWSTRIDE;
    float m = -3.4e38f;
    for (int c = lane; c < N; c += 32) m = fmaxf(m, row[c]);
    for (int mask = 16; mask >= 1; mask >>= 1)
      m = fmaxf(m, __shfl_xor(m, mask, 32));
    float s = 0.f;
    for (int c = lane; c < N; c += 32) {
      const float e = __expf(row[c] - m);
      row[c] = e;
      s += e;
    }
    for (int mask = 16; mask >= 1; mask >>= 1) s += __shfl_xor(s, mask, 32);
    if (lane == 0) s_inv[r] = 1.0f / s;
  }
  __syncthreads();

  // ---- Stage C: h_prime tile (16x32) = P @ h, K=2048 split across waves --
  v8f c0 = {};
  v8f c1 = {};
  const int kbase = wave * 256;
  for (int kk = 0; kk < 256; kk += 4) {
    const int k0 = kbase + kk;
    v2f a = *(const v2f*)(sc + nn * ROWSTRIDE + k0 + kd);  // M = row nn
    v2f b0, b1;
    b0.x = h[(size_t)(k0 + kd) * D_OUT + nn];
    b0.y = h[(size_t)(k0 + kd + 1) * D_OUT + nn];
    b1.x = h[(size_t)(k0 + kd) * D_OUT + 16 + nn];
    b1.y = h[(size_t)(k0 + kd + 1) * D_OUT + 16 + nn];
    c0 = wmma_f32_k4(a, b0, c0);
    c1 = wmma_f32_k4(a, b1, c1);
  }
  {
    const int row = (lane >> 4) << 3;
    float* pw = part + wave * 512;
    for (int v = 0; v < 8; ++v) {
      pw[(row + v) * 32 + nn] = c0[v];
      pw[(row + v) * 32 + 16 + nn] = c1[v];
    }
  }
  __syncthreads();
  for (int e = tid; e < 512; e += 256) {
    float s = 0.f;
    for (int w = 0; w < 8; ++w) s += part[w * 512 + e];
    const int row = e >> 5, col = e & 31;
    h_prime[(size_t)(i0 + row) * D_OUT + col] = s * s_inv[row];
  }
}

// ---------------------------------------------------------------------------
// BatchNorm1d (training mode), deterministic 2-stage reduction, no atomics.
// ---------------------------------------------------------------------------
__global__ void bn_partial_kernel(const float* __restrict__ hp,
                                  float* __restrict__ bp) {
  __shared__ float ls[8][32];
  __shared__ float lq[8][32];
  const int col = threadIdx.x & 31;
  const int seg = threadIdx.x >> 5;
  const int r0 = blockIdx.x * 32 + seg * 4;
  float s = 0.f, q = 0.f;
  for (int r = 0; r < 4; ++r) {
    const float v = hp[(size_t)(r0 + r) * D_OUT + col];
    s += v;
    q += v * v;
  }
  ls[seg][col] = s;
  lq[seg][col] = q;
  __syncthreads();
  if (threadIdx.x < 32) {
    float ss = 0.f, qq = 0.f;
    for (int w = 0; w < 8; ++w) {
      ss += ls[w][col];
      qq += lq[w][col];
    }
    bp[blockIdx.x * 64 + col] = ss;
    bp[blockIdx.x * 64 + 32 + col] = qq;
  }
}

__global__ void bn_stats_kernel(const float* __restrict__ bp,
                                float* __restrict__ ms) {
  const int col = threadIdx.x;  // 32 threads
  float s = 0.f, q = 0.f;
  for (int b = 0; b < 64; ++b) {
    s += bp[b * 64 + col];
    q += bp[b * 64 + 32 + col];
  }
  const float mean = s * (1.0f / (float)N);
  const float var = q * (1.0f / (float)N) - mean * mean;  // biased, as jnp.var
  ms[col] = mean;
  ms[32 + col] = rsqrtf(var + 1e-5f);
}

__global__ void bn_norm_kernel(const float* __restrict__ hp,
                               const float* __restrict__ ms,
                               const float* __restrict__ gamma,
                               const float* __restrict__ beta,
                               float* __restrict__ out) {
  const int idx = blockIdx.x * blockDim.x + threadIdx.x;  // 65536
  const int col = idx & 31;
  out[idx] = gamma[col] * (hp[idx] - ms[col]) * ms[32 + col] + beta[col];
}

// ---------------------------------------------------------------------------
extern "C" void kernel_launch(void* const* d_in, const int* in_sizes, int n_in,
                              void* d_out, int out_size, void* d_ws,
                              size_t ws_size, hipStream_t stream) {
  const float* input = (const float*)d_in[0];
  const float* edge_feat = (const float*)d_in[1];
  const int* adj = (const int*)d_in[2];
  const float* Wh = (const float*)d_in[3];
  const float* Wh1 = (const float*)d_in[4];
  const float* Wh2 = (const float*)d_in[5];
  const float* Wf = (const float*)d_in[6];
  const float* ah = (const float*)d_in[7];
  const float* af = (const float*)d_in[8];
  const float* bf = (const float*)d_in[9];
  const float* gamma = (const float*)d_in[10];
  const float* beta = (const float*)d_in[11];

  float* ws = (float*)d_ws;
  float* v1 = ws;             // 128
  float* v2 = ws + 128;       // 128
  float* s1 = ws + 256;       // 2048
  float* s2 = ws + 2304;      // 2048
  float* h = ws + 4352;       // 65536
  float* hp = ws + 69888;     // 65536
  float* bp = ws + 135424;    // 64*64
  float* ms = ws + 139520;    // 64
  float* out = (float*)d_out;

  prep_vec_kernel<<<1, 128, 0, stream>>>(Wh1, Wh2, ah, v1, v2);
  prep_s_kernel<<<8, 256, 0, stream>>>(input, v1, v2, s1, s2);
  h_gemm_kernel<<<128, 32, 0, stream>>>(input, Wh, h);
  const size_t smem = (size_t)(16 * ROWSTRIDE + 8 * 512) * sizeof(float);
  fused_attention_kernel<<<128, 256, smem, stream>>>(edge_feat, adj, Wf, af, bf,
                                                     s1, s2, h, hp);
  bn_partial_kernel<<<64, 256, 0, stream>>>(hp, bp);
  bn_stats_kernel<<<1, 32, 0, stream>>>(bp, ms);
  bn_norm_kernel<<<256, 256, 0, stream>>>(hp, ms, gamma, beta, out);
}